// QVLoraExpertRouter_63153199120805
// MI455X (gfx1250) — compile-verified
//
#include <hip/hip_runtime.h>
#include <hip/hip_bf16.h>
#include <cstddef>
#include <cstdint>

typedef __attribute__((ext_vector_type(2))) float v2f;
typedef __attribute__((ext_vector_type(4))) float v4f;
typedef __attribute__((ext_vector_type(8))) float v8f;
typedef __attribute__((ext_vector_type(4))) int   v4i;

// D = A(16x4) * B(4x16) + C, fp32 WMMA (CDNA5)
#define WMMA4(A, B, C) \
  __builtin_amdgcn_wmma_f32_16x16x4_f32(false, (A), false, (B), (short)0, (C), false, false)

constexpr int D  = 4096;
constexpr int E  = 8;
constexpr int R  = 16;
constexpr int OQ = 4096;
constexpr int OV = 1024;
constexpr int T  = 16384;
constexpr float SCALE = 2.0f;   // attn_lora_alpha / attn_lora_rank = 32/16
constexpr int WAVES = 4;        // waves per workgroup (1 wave == 16 tokens)
constexpr int LP = 132;         // padded LDS row stride (floats): conflict-free frag reads
constexpr int KC = 128;         // h K-chunk staged per async step
constexpr int NCH = D / KC;     // 32 chunks
constexpr int HP = 132;         // padded h-chunk row stride (floats)

// ---- CDNA5 async global->LDS path (ASYNCcnt) with portable fallback --------
#if __has_builtin(__builtin_amdgcn_global_load_async_to_lds_b128)
#define HAVE_ASYNC_LDS 1
#endif

typedef __attribute__((address_space(1))) v4i gv4i;   // global int4
typedef __attribute__((address_space(3))) v4i lv4i;   // LDS int4

__device__ __forceinline__ void async_copy_b128(const float* g, float* l) {
#ifdef HAVE_ASYNC_LDS
  __builtin_amdgcn_global_load_async_to_lds_b128(
      (gv4i*)(uintptr_t)g,
      (lv4i*)(uint32_t)(uintptr_t)l,   // low 32 bits of flat shared addr = LDS offset
      0, 0);
#else
  *(v4f*)l = *(const v4f*)g;   // coalesced b128 load + ds_store_b128 staging
#endif
}

__device__ __forceinline__ void wait_async_16() {
#ifdef HAVE_ASYNC_LDS
#if __has_builtin(__builtin_amdgcn_s_wait_asynccnt)
  __builtin_amdgcn_s_wait_asynccnt(16);
#else
  asm volatile("s_wait_asynccnt 0x10" ::: "memory");
#endif
#endif
  asm volatile("" ::: "memory");   // compiler barrier: pin LDS consumers after wait
}

__device__ __forceinline__ void wait_async_0() {
#ifdef HAVE_ASYNC_LDS
#if __has_builtin(__builtin_amdgcn_s_wait_asynccnt)
  __builtin_amdgcn_s_wait_asynccnt(0);
#else
  asm volatile("s_wait_asynccnt 0x0" ::: "memory");
#endif
#endif
  asm volatile("" ::: "memory");
}

// Stage one [16 token x 128 float] h chunk into padded LDS (16 x b128/lane ops)
__device__ __forceinline__ void stage_chunk(const float* gbase, float* buf, int lane) {
  #pragma unroll
  for (int r = 0; r < 16; r++)
    async_copy_b128(gbase + (size_t)r * D + lane * 4, buf + r * HP + lane * 4);
}

__global__ __launch_bounds__(WAVES * 32, 1)
void qv_lora_router_kernel(const float* __restrict__ h,
                           const float* __restrict__ rw,
                           const float* __restrict__ qa,
                           const float* __restrict__ qb,
                           const float* __restrict__ va,
                           const float* __restrict__ vb,
                           float* __restrict__ outq,
                           float* __restrict__ outv)
{
  const int lane = threadIdx.x & 31;
  const int wave = threadIdx.x >> 5;
  const int hl   = lane & 15;          // column / M-index within half-wave
  const int kup  = (lane & 16) >> 3;   // 0 or 2 : K offset of upper half-wave
  const int rup  = (lane & 16) >> 1;   // 0 or 8 : row offset in C/D tiles
  const int t0   = blockIdx.x * (WAVES * 16) + wave * 16;

  __shared__ float sH [WAVES][2][16 * HP];  // double-buffered h chunk per wave
  __shared__ float sLq[WAVES][16 * LP];     // selected+scaled q low-rank [16 x 128]
  __shared__ float sLv[WAVES][16 * LP];     // selected+scaled v low-rank
  __shared__ float sLg[WAVES][16 * E];      // router logits
  __shared__ float sSc[WAVES][16];          // score * SCALE
  __shared__ int   sIx[WAVES][16];          // argmax expert

  float* Lq = sLq[wave];
  float* Lv = sLv[wave];
  float* Lg = sLg[wave];

  // ---------------- Phase A: fused router + all-expert low-rank -------------
  // acc[0]     : router logits tile [16 x 16] (cols 0..7 valid)
  // acc[1..8]  : q low-rank tiles, expert e -> cols e*16..e*16+15
  // acc[9..16] : v low-rank tiles
  v8f acc[2 * E + 1];
  #pragma unroll
  for (int i = 0; i < 2 * E + 1; i++)
    acc[i] = v8f{0.f, 0.f, 0.f, 0.f, 0.f, 0.f, 0.f, 0.f};

  const float* hbase = h + (size_t)t0 * D;
  float* hb0 = sH[wave][0];
  float* hb1 = sH[wave][1];

  stage_chunk(hbase, hb0, lane);
  for (int c = 0; c < NCH; c++) {
    const float* cur = (c & 1) ? hb1 : hb0;
    if (c + 1 < NCH) {
      stage_chunk(hbase + (size_t)(c + 1) * KC, (c & 1) ? hb0 : hb1, lane);
      wait_async_16();   // 16 new in flight; chunk c's 16 completed (in-order)
    } else {
      wait_async_0();
    }
    for (int k = 0; k < KC; k += 4) {
      const int kk = k + kup;            // K offset inside chunk
      const int kg = c * KC + kk;        // global K for weight fragments
      v2f a;                             // A-fragment from LDS (ds_load_b64)
      a.x = cur[hl * HP + kk];
      a.y = cur[hl * HP + kk + 1];

      { // router: B[k][n] = rw[n*D + k], n < 8, zero-padded to 16
        v2f b;
        b.x = (hl < E) ? rw[(size_t)hl * D + kg]     : 0.f;
        b.y = (hl < E) ? rw[(size_t)hl * D + kg + 1] : 0.f;
        acc[0] = WMMA4(a, b, acc[0]);
      }
      #pragma unroll
      for (int e = 0; e < E; e++) {   // q_lora_a[e, k, r]: 64B contiguous per row
        const float* bb = qa + (size_t)e * D * R + (size_t)kg * R + hl;
        v2f b; b.x = bb[0]; b.y = bb[R];
        acc[1 + e] = WMMA4(a, b, acc[1 + e]);
      }
      #pragma unroll
      for (int e = 0; e < E; e++) {
        const float* bb = va + (size_t)e * D * R + (size_t)kg * R + hl;
        v2f b; b.x = bb[0]; b.y = bb[R];
        acc[1 + E + e] = WMMA4(a, b, acc[1 + E + e]);
      }
    }
  }

  // ---------------- router softmax / argmax ---------------------------------
  #pragma unroll
  for (int i = 0; i < 8; i++) {
    if (hl < E) Lg[(i + rup) * E + hl] = acc[0][i];
  }
  if (lane < 16) {   // token = lane; intra-wave LDS RAW handled by dscnt waits
    float m = Lg[lane * E];
    int bi = 0;
    #pragma unroll
    for (int e = 1; e < E; e++) {
      float v = Lg[lane * E + e];
      if (v > m) { m = v; bi = e; }
    }
    float s = 0.f;
    #pragma unroll
    for (int e = 0; e < E; e++) s += __expf(Lg[lane * E + e] - m);
    sIx[wave][lane] = bi;
    sSc[wave][lane] = SCALE / s;   // max prob = 1/sum(exp(l - max)), pre-scaled
  }

  // ---------------- zero-select + scale: build L matrices in LDS ------------
  // L[t, e*16 + r] = (e == idx[t]) ? lowrank[t, e*16+r] * score[t] * SCALE : 0
  #pragma unroll
  for (int e = 0; e < E; e++) {
    #pragma unroll
    for (int i = 0; i < 8; i++) {
      const int t = i + rup;
      const float sc = (sIx[wave][t] == e) ? sSc[wave][t] : 0.f;
      Lq[t * LP + e * 16 + hl] = acc[1 + e][i]     * sc;
      Lv[t * LP + e * 16 + hl] = acc[1 + E + e][i] * sc;
    }
  }

  // ---------------- Phase B: delta = L[16,128] @ lora_b_flat[128, O] --------
  { // q: O = 4096
    v2f af[32];
    #pragma unroll
    for (int kc = 0; kc < 32; kc++) {
      const int kk = kc * 4 + kup;
      af[kc].x = Lq[hl * LP + kk];
      af[kc].y = Lq[hl * LP + kk + 1];
    }
    float* oq = outq + (size_t)t0 * OQ;
    for (int n0 = 0; n0 < OQ; n0 += 16) {
      v8f c = v8f{0.f, 0.f, 0.f, 0.f, 0.f, 0.f, 0.f, 0.f};
      #pragma unroll
      for (int kc = 0; kc < 32; kc++) {
        const int kk = kc * 4 + kup;
        const float* bb = qb + (size_t)kk * OQ + n0 + hl;  // [128, 4096] row-major
        v2f b; b.x = bb[0]; b.y = bb[OQ];
        c = WMMA4(af[kc], b, c);
      }
      #pragma unroll
      for (int i = 0; i < 8; i++)
        oq[(size_t)(i + rup) * OQ + n0 + hl] = c[i];
    }
  }
  { // v: O = 1024
    v2f af[32];
    #pragma unroll
    for (int kc = 0; kc < 32; kc++) {
      const int kk = kc * 4 + kup;
      af[kc].x = Lv[hl * LP + kk];
      af[kc].y = Lv[hl * LP + kk + 1];
    }
    float* ov = outv + (size_t)t0 * OV;
    for (int n0 = 0; n0 < OV; n0 += 16) {
      v8f c = v8f{0.f, 0.f, 0.f, 0.f, 0.f, 0.f, 0.f, 0.f};
      #pragma unroll
      for (int kc = 0; kc < 32; kc++) {
        const int kk = kc * 4 + kup;
        const float* bb = vb + (size_t)kk * OV + n0 + hl;  // [128, 1024] row-major
        v2f b; b.x = bb[0]; b.y = bb[OV];
        c = WMMA4(af[kc], b, c);
      }
      #pragma unroll
      for (int i = 0; i < 8; i++)
        ov[(size_t)(i + rup) * OV + n0 + hl] = c[i];
    }
  }
}

extern "C" void kernel_launch(void* const* d_in, const int* in_sizes, int n_in,
                              void* d_out, int out_size, void* d_ws, size_t ws_size,
                              hipStream_t stream) {
  const float* h  = (const float*)d_in[0];   // [4,4096,4096]
  const float* rw = (const float*)d_in[1];   // [8,4096]
  const float* qa = (const float*)d_in[2];   // [8,4096,16]
  const float* qb = (const float*)d_in[3];   // [8,16,4096]
  const float* va = (const float*)d_in[4];   // [8,4096,16]
  const float* vb = (const float*)d_in[5];   // [8,16,1024]
  float* outq = (float*)d_out;               // [16384, 4096]
  float* outv = outq + (size_t)T * OQ;       // [16384, 1024]

  dim3 grid(T / (WAVES * 16));               // 256 workgroups
  dim3 block(WAVES * 32);                    // 4 wave32 per workgroup
  qv_lora_router_kernel<<<grid, block, 0, stream>>>(h, rw, qa, qb, va, vb, outq, outv);
}